// SAGE_11244224381113
// MI455X (gfx1250) — compile-verified
//
#include <hip/hip_runtime.h>
#include <hip/hip_bf16.h>

typedef __attribute__((ext_vector_type(2))) float v2f;
typedef __attribute__((ext_vector_type(8))) float v8f;

#define SAGE_N   100000
#define SAGE_E   1600000
#define SAGE_K   128      // input feature dim of every layer's GEMM
#define LDS_PITCH 130     // 128 + 2 pad: row stride in floats, kills 16-way bank conflicts

// ---------------------------------------------------------------------------
// Degree: deg[d] += 1 per edge, then deg_inv = 1/max(deg,1)
// ---------------------------------------------------------------------------
__global__ void __launch_bounds__(256)
sage_deg_kernel(const int* __restrict__ dst, float* __restrict__ deg, int n_edges) {
    int e = blockIdx.x * blockDim.x + threadIdx.x;
    if (e < n_edges) atomicAdd(&deg[dst[e]], 1.0f);
}

__global__ void __launch_bounds__(256)
sage_deginv_kernel(float* __restrict__ deg, int n) {
    int i = blockIdx.x * blockDim.x + threadIdx.x;
    if (i < n) deg[i] = 1.0f / fmaxf(deg[i], 1.0f);
}

// ---------------------------------------------------------------------------
// Edge-parallel neighbor sum: 32 threads per edge, float4 per thread.
// agg[dst[e]][:] += h[src[e]][:]   (scaling by deg_inv is folded into GEMM)
// Gather is 512B contiguous per edge; h and agg stay L2-resident (51 MB each).
// ---------------------------------------------------------------------------
__global__ void __launch_bounds__(256)
sage_aggregate_kernel(const float* __restrict__ h, const int* __restrict__ src,
                      const int* __restrict__ dst, float* __restrict__ agg,
                      int n_edges) {
    const int tpe = SAGE_K / 4;                      // 32 threads per edge
    long long gid = (long long)blockIdx.x * blockDim.x + threadIdx.x;
    int e = (int)(gid / tpe);
    int c = (int)(gid % tpe);
    if (e >= n_edges) return;
    int s = src[e];
    int t = dst[e];
    const float4 v = ((const float4*)(h + (size_t)s * SAGE_K))[c];
    float* p = agg + (size_t)t * SAGE_K + (size_t)c * 4;
    atomicAdd(p + 0, v.x);
    atomicAdd(p + 1, v.y);
    atomicAdd(p + 2, v.z);
    atomicAdd(p + 3, v.w);
}

// ---------------------------------------------------------------------------
// Fused dual GEMM + bias + (optional) ReLU, full fp32 via V_WMMA_F32_16X16X4_F32.
//   out = act( h @ Ws + (agg * dinv[:,None]) @ Wn + b )
// One block owns one 16-row M-tile; one wave per 16-col N-tile
// (blockDim.x = 32 * dout/16: 256 threads for dout=128, 128 for dout=64).
// The A-tile (h and deg-scaled agg, 16x128 each) is staged ONCE in LDS and
// shared by all N-tile waves, cutting A-side L2 traffic by 8x. Grid geometry
// is exact (100000 = 6250*16), so no wave diverges: EXEC is all-ones at every
// WMMA and every wave reaches the barrier.
// ---------------------------------------------------------------------------
__global__ void __launch_bounds__(256)
sage_gemm_kernel(const float* __restrict__ h, const float* __restrict__ agg,
                 const float* __restrict__ dinv,
                 const float* __restrict__ Ws, const float* __restrict__ Wn,
                 const float* __restrict__ bias,
                 float* __restrict__ out, int dout, int relu) {
    __shared__ float lds_h[16 * LDS_PITCH];
    __shared__ float lds_g[16 * LDS_PITCH];

    const int tid = threadIdx.x;
    const int m0  = blockIdx.x * 16;

    // Cooperative stage of the A-tile: 16 rows x 128 cols per matrix,
    // float4 granularity, deg_inv folded into the neighbor matrix.
    const int nvec = 16 * (SAGE_K / 4);              // 512 float4 per matrix
    for (int i = tid; i < nvec; i += blockDim.x) {
        const int r  = i >> 5;                       // row 0..15
        const int c4 = i & 31;                       // float4 column 0..31
        const float di = dinv[m0 + r];
        const float4 v = ((const float4*)(h   + (size_t)(m0 + r) * SAGE_K))[c4];
        const float4 g = ((const float4*)(agg + (size_t)(m0 + r) * SAGE_K))[c4];
        float* dh = &lds_h[r * LDS_PITCH + c4 * 4];
        float* dg = &lds_g[r * LDS_PITCH + c4 * 4];
        dh[0] = v.x;      dh[1] = v.y;      dh[2] = v.z;      dh[3] = v.w;
        dg[0] = g.x * di; dg[1] = g.y * di; dg[2] = g.z * di; dg[3] = g.w * di;
    }
    __syncthreads();

    const int lane   = tid & 31;
    const int tile_n = tid >> 5;                     // wave id == N-tile
    const int n0     = tile_n * 16;
    const int lm     = lane & 15;                    // A row / B,C column in tile
    const int hi     = lane >> 4;                    // lane half: K pair / M half

    const float* hrow = &lds_h[lm * LDS_PITCH];
    const float* grow = &lds_g[lm * LDS_PITCH];

    v8f c = {};
    #pragma unroll 4
    for (int k = 0; k < SAGE_K; k += 4) {
        const int ka = k + hi * 2;
        // A tiles (16x4 f32): lanes 0-15 hold K=k,k+1 ; lanes 16-31 hold K=k+2,k+3
        v2f a_self  = { hrow[ka], hrow[ka + 1] };
        v2f a_neigh = { grow[ka], grow[ka + 1] };
        // B tiles (4x16 f32): VGPR j holds row (k+j) for lanes 0-15, row (k+j+2)
        // for lanes 16-31, column n0+lm.
        const float* wsp = Ws + (size_t)ka * dout + n0 + lm;
        const float* wnp = Wn + (size_t)ka * dout + n0 + lm;
        v2f b_self  = { wsp[0], wsp[dout] };
        v2f b_neigh = { wnp[0], wnp[dout] };
        c = __builtin_amdgcn_wmma_f32_16x16x4_f32(false, a_self,  false, b_self,
                                                  (short)0, c, false, false);
        c = __builtin_amdgcn_wmma_f32_16x16x4_f32(false, a_neigh, false, b_neigh,
                                                  (short)0, c, false, false);
    }

    // Epilogue: C layout — VGPR j: M = m0 + j (+8 for lanes 16-31), N = n0 + lm.
    const float bn = bias[n0 + lm];
    #pragma unroll
    for (int j = 0; j < 8; ++j) {
        float v = c[j] + bn;
        if (relu) v = fmaxf(v, 0.0f);
        out[(size_t)(m0 + hi * 8 + j) * dout + n0 + lm] = v;
    }
}

// ---------------------------------------------------------------------------
// Launch: 3 SAGE layers. Workspace layout (floats):
//   bufA [0, ND)       : neighbor-sum accumulator (re-zeroed per layer)
//   bufB [ND, 2ND)     : h1
//   bufC [2ND, 3ND)    : h2
//   deg  [3ND, 3ND+N)  : degree -> deg_inv
// ---------------------------------------------------------------------------
extern "C" void kernel_launch(void* const* d_in, const int* in_sizes, int n_in,
                              void* d_out, int out_size, void* d_ws, size_t ws_size,
                              hipStream_t stream) {
    (void)in_sizes; (void)n_in; (void)out_size; (void)ws_size;

    const float* x   = (const float*)d_in[0];
    const int*   src = (const int*)  d_in[1];
    const int*   dst = (const int*)  d_in[2];
    const float* Ws0 = (const float*)d_in[3];
    const float* Wn0 = (const float*)d_in[4];
    const float* b0  = (const float*)d_in[5];
    const float* Ws1 = (const float*)d_in[6];
    const float* Wn1 = (const float*)d_in[7];
    const float* b1  = (const float*)d_in[8];
    const float* Ws2 = (const float*)d_in[9];
    const float* Wn2 = (const float*)d_in[10];
    const float* b2  = (const float*)d_in[11];
    float* out = (float*)d_out;

    const size_t ND = (size_t)SAGE_N * SAGE_K;     // 12,800,000 floats
    float* bufA = (float*)d_ws;                    // aggregation accumulator
    float* bufB = bufA + ND;                       // h1
    float* bufC = bufB + ND;                       // h2
    float* deg  = bufC + ND;                       // N floats

    const int aggBlocks = (int)(((long long)SAGE_E * (SAGE_K / 4) + 255) / 256);
    const int degBlocks = (SAGE_E + 255) / 256;
    const int invBlocks = (SAGE_N + 255) / 256;
    const int mTiles    = SAGE_N / 16;             // 6250, exact

    // --- degree / deg_inv (shared across layers) ---
    hipMemsetAsync(deg, 0, (size_t)SAGE_N * sizeof(float), stream);
    sage_deg_kernel<<<degBlocks, 256, 0, stream>>>(dst, deg, SAGE_E);
    sage_deginv_kernel<<<invBlocks, 256, 0, stream>>>(deg, SAGE_N);

    // --- layer 0: x -> bufB (ReLU), dout=128 -> 8 waves/block ---
    hipMemsetAsync(bufA, 0, ND * sizeof(float), stream);
    sage_aggregate_kernel<<<aggBlocks, 256, 0, stream>>>(x, src, dst, bufA, SAGE_E);
    sage_gemm_kernel<<<mTiles, 32 * (128 / 16), 0, stream>>>(
        x, bufA, deg, Ws0, Wn0, b0, bufB, 128, 1);

    // --- layer 1: bufB -> bufC (ReLU), dout=128 ---
    hipMemsetAsync(bufA, 0, ND * sizeof(float), stream);
    sage_aggregate_kernel<<<aggBlocks, 256, 0, stream>>>(bufB, src, dst, bufA, SAGE_E);
    sage_gemm_kernel<<<mTiles, 32 * (128 / 16), 0, stream>>>(
        bufB, bufA, deg, Ws1, Wn1, b1, bufC, 128, 1);

    // --- layer 2: bufC -> out (no ReLU), dout=64 -> 4 waves/block ---
    hipMemsetAsync(bufA, 0, ND * sizeof(float), stream);
    sage_aggregate_kernel<<<aggBlocks, 256, 0, stream>>>(bufC, src, dst, bufA, SAGE_E);
    sage_gemm_kernel<<<mTiles, 32 * (64 / 16), 0, stream>>>(
        bufC, bufA, deg, Ws2, Wn2, b2, out, 64, 0);
}